// DCTM_24223615549521
// MI455X (gfx1250) — compile-verified
//
#include <hip/hip_runtime.h>
#include <hip/hip_bf16.h>
#include <math.h>

// ---------------------------------------------------------------------------
// Problem constants (from reference): B=4, C=192, H=W=64, heads=6, hid=768
// ---------------------------------------------------------------------------
#define BATCH 4
#define CH    192
#define HH    64
#define WW    64
#define HWSZ  4096          // H*W
#define HEADS 6
#define HDIM  32            // C / heads
#define HID   768           // 4*C
#define NTAP  9             // KS*KS
#define LN_EPS 1e-5f

// GEMM tiling for the LDS-staged kernel
#define BM 64
#define BN 64
#define BK 16
#define APAD 2              // A-tile row pad (keeps v2f reads 8B aligned, spreads banks)

typedef __attribute__((ext_vector_type(2))) float v2f;
typedef __attribute__((ext_vector_type(8))) float v8f;

// ---------------------------------------------------------------------------
// LayerNorm over channel axis; tensors are channel-major: t[b][c][ij]
// ---------------------------------------------------------------------------
__global__ __launch_bounds__(256)
void ln_kernel(const float* __restrict__ in, const float* __restrict__ g,
               const float* __restrict__ be, float* __restrict__ out)
{
    int gid = blockIdx.x * 256 + threadIdx.x;
    if (gid >= BATCH * HWSZ) return;
    int b = gid >> 12;
    int ij = gid & (HWSZ - 1);
    const float* p = in + (long long)b * CH * HWSZ + ij;
    float s = 0.f;
    for (int c = 0; c < CH; ++c) s += p[(long long)c * HWSZ];
    float mu = s * (1.0f / CH);
    float v = 0.f;
    for (int c = 0; c < CH; ++c) {
        float d = p[(long long)c * HWSZ] - mu;
        v += d * d;
    }
    float inv = rsqrtf(v * (1.0f / CH) + LN_EPS);
    float* o = out + (long long)b * CH * HWSZ + ij;
    for (int c = 0; c < CH; ++c) {
        float d = (p[(long long)c * HWSZ] - mu) * inv;
        o[(long long)c * HWSZ] = d * g[c] + be[c];
    }
}

// ---------------------------------------------------------------------------
// Offset conv (18ch 3x3 pad1) + bilinear sampling metadata.
// meta layout: [b][tap][corner][ij]
// ---------------------------------------------------------------------------
__global__ __launch_bounds__(256)
void conv_off_kernel(const float* __restrict__ xn, const float* __restrict__ w_p,
                     const float* __restrict__ b_p,
                     int* __restrict__ meti, float* __restrict__ metw)
{
    int gid = blockIdx.x * 256 + threadIdx.x;
    if (gid >= BATCH * HWSZ) return;
    int b = gid >> 12;
    int ij = gid & (HWSZ - 1);
    int i = ij >> 6, j = ij & 63;

    float acc[18];
#pragma unroll
    for (int o = 0; o < 18; ++o) acc[o] = 0.f;

    for (int c = 0; c < CH; ++c) {
        const float* xc = xn + ((long long)b * CH + c) * HWSZ;
        float nb[9];
#pragma unroll
        for (int uu = 0; uu < 3; ++uu) {
#pragma unroll
            for (int vv = 0; vv < 3; ++vv) {
                int ii = i + uu - 1, jj = j + vv - 1;
                nb[uu * 3 + vv] =
                    (ii >= 0 && ii < HH && jj >= 0 && jj < WW) ? xc[ii * WW + jj] : 0.f;
            }
        }
#pragma unroll
        for (int o = 0; o < 18; ++o) {
            const float* w = w_p + ((long long)o * CH + c) * 9;
            float s = acc[o];
#pragma unroll
            for (int t = 0; t < 9; ++t) s += w[t] * nb[t];
            acc[o] = s;
        }
    }
#pragma unroll
    for (int o = 0; o < 18; ++o) acc[o] += b_p[o];

#pragma unroll
    for (int n = 0; n < NTAP; ++n) {
        float dx = (float)(n / 3 - 1);
        float dy = (float)(n % 3 - 1);
        float px = acc[n]     + (float)(i + 1) + dx;
        float py = acc[9 + n] + (float)(j + 1) + dy;
        float fx = floorf(px), fy = floorf(py);
        float qlx = fminf(fmaxf(fx,       0.f), 65.f);
        float qly = fminf(fmaxf(fy,       0.f), 65.f);
        float qrx = fminf(fmaxf(fx + 1.f, 0.f), 65.f);
        float qry = fminf(fmaxf(fy + 1.f, 0.f), 65.f);
        float pxc = fminf(fmaxf(px, 0.f), 65.f);
        float pyc = fminf(fmaxf(py, 0.f), 65.f);
        float glt = (1.f + (qlx - pxc)) * (1.f + (qly - pyc));
        float grb = (1.f - (qrx - pxc)) * (1.f - (qry - pyc));
        float glb = (1.f + (qlx - pxc)) * (1.f - (qry - pyc));
        float grt = (1.f - (qrx - pxc)) * (1.f + (qly - pyc));

        float cx[4] = {qlx, qrx, qlx, qrx};
        float cy[4] = {qly, qry, qry, qly};
        float cw[4] = {glt, grb, glb, grt};
        long long mbase = (((long long)b * NTAP + n) * 4) * HWSZ + ij;
#pragma unroll
        for (int t = 0; t < 4; ++t) {
            int xi = (int)cx[t], yi = (int)cy[t];
            int idx = (xi >= 1 && xi <= HH && yi >= 1 && yi <= WW)
                          ? (xi - 1) * WW + (yi - 1) : -1;
            meti[mbase + (long long)t * HWSZ] = idx;
            metw[mbase + (long long)t * HWSZ] = cw[t];
        }
    }
}

// ---------------------------------------------------------------------------
// Build sampled matrix S[c*9+n][ij] for one batch: 4-corner bilinear gather.
// ---------------------------------------------------------------------------
__global__ __launch_bounds__(256)
void build_s_kernel(const float* __restrict__ xn, const int* __restrict__ meti,
                    const float* __restrict__ metw, float* __restrict__ S, int b)
{
    long long r = (long long)blockIdx.x * 256 + threadIdx.x;
    if (r >= (long long)CH * NTAP * HWSZ) return;
    int cn = (int)(r >> 12);
    int ij = (int)(r & (HWSZ - 1));
    int c = cn / NTAP;
    int n = cn - c * NTAP;
    const float* xc = xn + ((long long)b * CH + c) * HWSZ;
    long long mbase = (((long long)b * NTAP + n) * 4) * HWSZ + ij;
    float s = 0.f;
#pragma unroll
    for (int t = 0; t < 4; ++t) {
        int idx = meti[mbase + (long long)t * HWSZ];
        float w = metw[mbase + (long long)t * HWSZ];
        if (idx >= 0) s += w * xc[idx];
    }
    S[r] = s;
}

// ---------------------------------------------------------------------------
// L2-normalize each row of length HWSZ in place (block per row, LDS reduce).
// ---------------------------------------------------------------------------
__global__ __launch_bounds__(256)
void l2norm_kernel(float* __restrict__ p)
{
    __shared__ float red[256];
    long long base = (long long)blockIdx.x * HWSZ;
    float ss = 0.f;
    for (int e = threadIdx.x; e < HWSZ; e += 256) {
        float v = p[base + e];
        ss += v * v;
    }
    red[threadIdx.x] = ss;
    __syncthreads();
    for (int s = 128; s > 0; s >>= 1) {
        if (threadIdx.x < s) red[threadIdx.x] += red[threadIdx.x + s];
        __syncthreads();
    }
    float inv = 1.0f / fmaxf(sqrtf(red[0]), 1e-12f);
    for (int e = threadIdx.x; e < HWSZ; e += 256)
        p[base + e] *= inv;
}

// ---------------------------------------------------------------------------
// Temperature-scaled softmax over last dim (32). One thread per row.
// ---------------------------------------------------------------------------
__global__ __launch_bounds__(256)
void softmax_kernel(float* __restrict__ attn, const float* __restrict__ temp)
{
    int t = blockIdx.x * 256 + threadIdx.x;
    if (t >= BATCH * HEADS * HDIM) return;
    int bh = t / HDIM;
    int row = t - bh * HDIM;
    float tv = temp[bh % HEADS];
    float* a = attn + (long long)bh * HDIM * HDIM + (long long)row * HDIM;
    float m = -3.402823466e38f;
    float buf[HDIM];
#pragma unroll
    for (int c = 0; c < HDIM; ++c) {
        buf[c] = a[c] * tv;
        m = fmaxf(m, buf[c]);
    }
    float s = 0.f;
#pragma unroll
    for (int c = 0; c < HDIM; ++c) {
        buf[c] = __expf(buf[c] - m);
        s += buf[c];
    }
    float inv = 1.0f / s;
#pragma unroll
    for (int c = 0; c < HDIM; ++c) a[c] = buf[c] * inv;
}

// ---------------------------------------------------------------------------
// Epilogue helper: 0 none | 1 +res | 2 +bias,gelu | 3 +bias+res
// ---------------------------------------------------------------------------
__device__ __forceinline__ float apply_epi(float v, int epi, int m, int off,
                                           const float* bias, const float* res)
{
    if (epi == 1) {
        v += res[off];
    } else if (epi == 2) {
        v += bias[m];
        v = 0.5f * v * (1.0f + erff(v * 0.70710678118654752f));
    } else if (epi == 3) {
        v += bias[m] + res[off];
    }
    return v;
}

// ---------------------------------------------------------------------------
// LDS-staged, double-buffered, software-pipelined fp32 WMMA GEMM.
//   D[m][n] = sum_k A[m][k]*B[k][n]   (+epilogue)
// Requirements: M % 64 == 0, N % 64 == 0, K % 16 == 0, B has n contiguous
// (row stride ldB). A strides arbitrary (handles transposed weights free).
// Block: 256 threads / 8 waves; block tile 64x64x16; each wave: 32(M)x16(N)
// via two v8f accumulators sharing one B fragment -> 8 WMMA per K-step.
// fp32 frag layouts (ISA 7.12.2):
//   A: VGPR v, lane l -> M=l%16, K=v+2*(l/16)
//   B: VGPR v, lane l -> K=v+2*(l/16), N=l%16
// ---------------------------------------------------------------------------
__global__ __launch_bounds__(256)
void gemm_f32_wmma_lds(const float* __restrict__ A, int sAm, int sAk, long long bA,
                       const float* __restrict__ Bm, int ldB, long long bB,
                       float* __restrict__ Cm, int ldC, long long bC,
                       int M, int N, int K,
                       int epi, const float* __restrict__ bias,
                       const float* __restrict__ res, long long bRes)
{
    __shared__ float As[2][BM][BK + APAD];
    __shared__ float Bs[2][BK][BN];

    int z = blockIdx.z;
    A  += bA * z;
    Bm += bB * z;
    Cm += bC * z;
    if (res) res += bRes * z;

    int tid  = threadIdx.x;
    int wave = tid >> 5;
    int lane = tid & 31;
    int lo   = lane & 15;
    int hi   = lane >> 4;
    int wm   = wave >> 2;          // 0..1 : 32-row slab
    int wn   = wave & 3;           // 0..3 : 16-col slab

    int mblk = blockIdx.y * BM;
    int nblk = blockIdx.x * BN;

    // A-tile loader: thread t loads 4 elements (m = (t>>4)+i*16, kk = t&15)
    int a_m  = tid >> 4;           // 0..15 (+i*16)
    int a_kk = tid & 15;
    // B-tile loader: one float4 per thread (kk = t>>4, nn = (t&15)*4)
    int b_kk = tid >> 4;
    int b_nn = (tid & 15) * 4;

    float  ra[4];
    float4 rb;

    auto fetch = [&](int kbase) {
#pragma unroll
        for (int i = 0; i < 4; ++i)
            ra[i] = A[(mblk + a_m + i * 16) * sAm + (kbase + a_kk) * sAk];
        rb = *(const float4*)&Bm[(kbase + b_kk) * ldB + nblk + b_nn];
    };
    auto stage = [&](int buf) {
#pragma unroll
        for (int i = 0; i < 4; ++i)
            As[buf][a_m + i * 16][a_kk] = ra[i];
        *(float4*)&Bs[buf][b_kk][b_nn] = rb;
    };

    v8f acc0 = {};
    v8f acc1 = {};

    fetch(0);
    stage(0);
    __syncthreads();

    int nsteps = K / BK;
    for (int t = 0; t < nsteps; ++t) {
        int cur = t & 1;
        int nxt = cur ^ 1;
        if (t + 1 < nsteps) fetch((t + 1) * BK);   // loads in flight over compute

#pragma unroll
        for (int ks = 0; ks < BK; ks += 4) {
            int kk = ks + 2 * hi;
            v2f a0 = *(const v2f*)&As[cur][wm * 32 + lo][kk];
            v2f a1 = *(const v2f*)&As[cur][wm * 32 + 16 + lo][kk];
            v2f b;
            b.x = Bs[cur][kk][wn * 16 + lo];
            b.y = Bs[cur][kk + 1][wn * 16 + lo];
            acc0 = __builtin_amdgcn_wmma_f32_16x16x4_f32(
                false, a0, false, b, (short)0, acc0, false, false);
            acc1 = __builtin_amdgcn_wmma_f32_16x16x4_f32(
                false, a1, false, b, (short)0, acc1, false, false);
        }

        if (t + 1 < nsteps) stage(nxt);            // waits loads after compute
        __syncthreads();
    }

    int n = nblk + wn * 16 + lo;
#pragma unroll
    for (int r = 0; r < 8; ++r) {
        int m0 = mblk + wm * 32 + r + 8 * hi;
        int m1 = m0 + 16;
        int off0 = m0 * ldC + n;
        int off1 = m1 * ldC + n;
        Cm[off0] = apply_epi(acc0[r], epi, m0, off0, bias, res);
        Cm[off1] = apply_epi(acc1[r], epi, m1, off1, bias, res);
    }
}

// ---------------------------------------------------------------------------
// Generic strided WMMA GEMM (kept for the small attention GEMMs: M=32).
// ---------------------------------------------------------------------------
__global__ __launch_bounds__(128)
void gemm_f32_wmma_gen(const float* __restrict__ A, long long sAm, long long sAk, long long bA,
                       const float* __restrict__ Bm, long long sBk, long long sBn, long long bB,
                       float* __restrict__ Cm, long long ldC, long long bC,
                       int M, int N, int K)
{
    int z = blockIdx.z;
    A  += bA * z;
    Bm += bB * z;
    Cm += bC * z;

    int wave = threadIdx.x >> 5;
    int lane = threadIdx.x & 31;
    int lo = lane & 15;
    int hi = lane >> 4;
    int m0 = blockIdx.y * 32 + (wave >> 1) * 16;
    int n0 = blockIdx.x * 32 + (wave & 1) * 16;
    if (m0 >= M || n0 >= N) return;          // wave-uniform guard

    const float* ap = A + (long long)(m0 + lo) * sAm + 2LL * hi * sAk;
    const float* bp = Bm + (long long)(n0 + lo) * sBn + 2LL * hi * sBk;

    v8f acc = {};
#pragma unroll 4
    for (int k = 0; k < K; k += 4) {
        v2f a, b;
        a.x = ap[0];
        a.y = ap[sAk];
        b.x = bp[0];
        b.y = bp[sBk];
        acc = __builtin_amdgcn_wmma_f32_16x16x4_f32(
            false, a, false, b, (short)0, acc, false, false);
        ap += 4 * sAk;
        bp += 4 * sBk;
    }

#pragma unroll
    for (int r = 0; r < 8; ++r) {
        int m = m0 + r + 8 * hi;
        int n = n0 + lo;
        if (m < M && n < N)
            Cm[(long long)m * ldC + n] = acc[r];
    }
}

// ---------------------------------------------------------------------------
// Host orchestration
// ---------------------------------------------------------------------------
extern "C" void kernel_launch(void* const* d_in, const int* in_sizes, int n_in,
                              void* d_out, int out_size, void* d_ws, size_t ws_size,
                              hipStream_t stream)
{
    const float* x      = (const float*)d_in[0];
    const float* gamma1 = (const float*)d_in[1];
    const float* beta1  = (const float*)d_in[2];
    const float* gamma2 = (const float*)d_in[3];
    const float* beta2  = (const float*)d_in[4];
    const float* temp   = (const float*)d_in[5];
    const float* w_qkv  = (const float*)d_in[6];
    const float* w_proj = (const float*)d_in[7];
    const float* w_p    = (const float*)d_in[8];
    const float* b_p    = (const float*)d_in[9];
    const float* w_d    = (const float*)d_in[10];
    const float* w_fc1  = (const float*)d_in[11];
    const float* b_fc1  = (const float*)d_in[12];
    const float* w_fc2  = (const float*)d_in[13];
    const float* b_fc2  = (const float*)d_in[14];
    float* out = (float*)d_out;

    const long long FEAT  = (long long)BATCH * CH * HWSZ;        // 3,145,728 floats
    const long long METAN = (long long)BATCH * NTAP * 4 * HWSZ;  // 589,824

    float* ws   = (float*)d_ws;
    float* xn   = ws;
    float* kbuf = xn   + FEAT;
    float* vbuf = kbuf + FEAT;
    float* qbuf = vbuf + FEAT;
    float* xs2  = qbuf + FEAT;
    float* h1n  = xs2  + FEAT;
    float* aout = h1n  + FEAT;
    float* metw = aout + FEAT;
    int*   meti = (int*)(metw + METAN);
    float* attn = (float*)(meti + METAN);                        // 24,576 floats
    float* scratch = attn + (long long)BATCH * HEADS * HDIM * HDIM; // S / hidden

    const long long CHW = (long long)CH * HWSZ;

    // 1) LN1: x -> xn
    ln_kernel<<<(BATCH * HWSZ + 255) / 256, 256, 0, stream>>>(x, gamma1, beta1, xn);

    // 2) offsets + bilinear metadata
    conv_off_kernel<<<(BATCH * HWSZ + 255) / 256, 256, 0, stream>>>(
        xn, w_p, b_p, meti, metw);

    // 3) K = w_qkv[C:2C] @ xn ; V = w_qkv[2C:3C] @ xn   (batched over B)
    gemm_f32_wmma_lds<<<dim3(HWSZ / BN, CH / BM, BATCH), 256, 0, stream>>>(
        w_qkv + (long long)CH * CH, CH, 1, 0,
        xn, HWSZ, CHW,
        kbuf, HWSZ, CHW, CH, HWSZ, CH, 0, nullptr, nullptr, 0);
    gemm_f32_wmma_lds<<<dim3(HWSZ / BN, CH / BM, BATCH), 256, 0, stream>>>(
        w_qkv + 2LL * CH * CH, CH, 1, 0,
        xn, HWSZ, CHW,
        vbuf, HWSZ, CHW, CH, HWSZ, CH, 0, nullptr, nullptr, 0);

    // 4) l2norm rows of k-hat / v-hat
    l2norm_kernel<<<BATCH * CH, 256, 0, stream>>>(kbuf);
    l2norm_kernel<<<BATCH * CH, 256, 0, stream>>>(vbuf);

    // 5) deformable conv: gather S, then q = w_d(192x1728) @ S(1728x4096)
    for (int b = 0; b < BATCH; ++b) {
        long long stot = (long long)CH * NTAP * HWSZ;
        build_s_kernel<<<(unsigned)((stot + 255) / 256), 256, 0, stream>>>(
            xn, meti, metw, scratch, b);
        gemm_f32_wmma_lds<<<dim3(HWSZ / BN, CH / BM, 1), 256, 0, stream>>>(
            w_d, CH * NTAP, 1, 0,
            scratch, HWSZ, 0,
            qbuf + (long long)b * CHW, HWSZ, 0,
            CH, HWSZ, CH * NTAP, 0, nullptr, nullptr, 0);
    }

    // 6) attn = qh @ kh^T  (per b,head: 32x32, K=4096)
    gemm_f32_wmma_gen<<<dim3(1, 1, BATCH * HEADS), 128, 0, stream>>>(
        qbuf, HWSZ, 1, (long long)HDIM * HWSZ,
        kbuf, 1, HWSZ, (long long)HDIM * HWSZ,
        attn, HDIM, (long long)HDIM * HDIM,
        HDIM, HDIM, HWSZ);

    // 7) temperature-scaled softmax
    softmax_kernel<<<(BATCH * HEADS * HDIM + 255) / 256, 256, 0, stream>>>(attn, temp);

    // 8) out = attn @ vh  (32x4096, K=32)
    gemm_f32_wmma_gen<<<dim3(HWSZ / 32, 1, BATCH * HEADS), 128, 0, stream>>>(
        attn, HDIM, 1, (long long)HDIM * HDIM,
        vbuf, HWSZ, 1, (long long)HDIM * HWSZ,
        aout, HWSZ, (long long)HDIM * HWSZ,
        HDIM, HWSZ, HDIM);

    // 9) xs2 = shortcut(x) + w_proj @ aout
    gemm_f32_wmma_lds<<<dim3(HWSZ / BN, CH / BM, BATCH), 256, 0, stream>>>(
        w_proj, CH, 1, 0,
        aout, HWSZ, CHW,
        xs2, HWSZ, CHW, CH, HWSZ, CH, 1, nullptr, x, CHW);

    // 10) LN2: xs2 -> h1n
    ln_kernel<<<(BATCH * HWSZ + 255) / 256, 256, 0, stream>>>(xs2, gamma2, beta2, h1n);

    // 11) MLP per batch (hidden reuses scratch)
    for (int b = 0; b < BATCH; ++b) {
        gemm_f32_wmma_lds<<<dim3(HWSZ / BN, HID / BM, 1), 256, 0, stream>>>(
            w_fc1, 1, HID, 0,                       // A[m][c] = w_fc1[c][m]
            h1n + (long long)b * CHW, HWSZ, 0,
            scratch, HWSZ, 0,
            HID, HWSZ, CH, 2, b_fc1, nullptr, 0);
        gemm_f32_wmma_lds<<<dim3(HWSZ / BN, CH / BM, 1), 256, 0, stream>>>(
            w_fc2, 1, CH, 0,                        // A[o][m] = w_fc2[m][o]
            scratch, HWSZ, 0,
            out + (long long)b * CHW, HWSZ, 0,
            CH, HWSZ, HID, 3, b_fc2, xs2 + (long long)b * CHW, 0);
    }
}